// Model_17188459119223
// MI455X (gfx1250) — compile-verified
//
#include <hip/hip_runtime.h>
#include <math.h>

typedef __attribute__((ext_vector_type(2))) float v2f;
typedef __attribute__((ext_vector_type(8))) float v8f;

#define IMG_H 128
#define IMG_W 128
#define NFACE 1280
#define TSZ 4
#define NEAR_P 0.1f
#define FAR_P 100.0f
#define EPS_P 1e-8f

// ---------------------------------------------------------------------------
// Kernel 1: per-face affine coefficients.
// w1, w2, depth are affine in (px,py):  q = a*px + b*py + c
// Stored as float4 (a,b,c,0) per face so one b64 load per lane feeds the
// WMMA B-operand (lanes 0-15: K=0,1 -> (a,b); lanes 16-31: K=2,3 -> (c,0)).
// Degenerate faces (|det|<=EPS) get w1 == -1 so the inside test always fails.
// ---------------------------------------------------------------------------
__global__ void face_setup(const float* __restrict__ verts,
                           const int* __restrict__ faces,
                           float* __restrict__ cW1,
                           float* __restrict__ cW2,
                           float* __restrict__ cD,
                           float ex, float ey, float ez,
                           float xx, float xy, float xz,
                           float yx, float yy, float yz,
                           float zx, float zy, float zz) {
  int f = blockIdx.x * blockDim.x + threadIdx.x;
  if (f >= NFACE) return;
  float X[3], Y[3], Z[3];
#pragma unroll
  for (int i = 0; i < 3; ++i) {
    int vi = faces[f * 3 + i];
    float vx = verts[vi * 3 + 0] - ex;
    float vy = verts[vi * 3 + 1] - ey;
    float vz = verts[vi * 3 + 2] - ez;
    X[i] = xx * vx + xy * vy + xz * vz;   // cam-space x = dot(xax, v-eye)
    Y[i] = yx * vx + yy * vy + yz * vz;
    Z[i] = zx * vx + zy * vy + zz * vz;
  }
  float e1x = X[1] - X[0], e1y = Y[1] - Y[0];
  float e2x = X[2] - X[0], e2y = Y[2] - Y[0];
  float det = e1x * e2y - e2x * e1y;
  bool good = fabsf(det) > EPS_P;
  float inv = 1.0f / (good ? det : 1.0f);
  // w1 = (dx*e2y - dy*e2x)*inv,  dx = px - x0, dy = py - y0
  float a1 = e2y * inv, b1 = -e2x * inv, c1 = (Y[0] * e2x - X[0] * e2y) * inv;
  // w2 = (e1x*dy - e1y*dx)*inv
  float a2 = -e1y * inv, b2 = e1x * inv, c2 = (X[0] * e1y - Y[0] * e1x) * inv;
  // depth = z0 + w1*(z1-z0) + w2*(z2-z0)
  float dz1 = Z[1] - Z[0], dz2 = Z[2] - Z[0];
  float ad = a1 * dz1 + a2 * dz2;
  float bd = b1 * dz1 + b2 * dz2;
  float cd = Z[0] + c1 * dz1 + c2 * dz2;
  if (!good) { a1 = 0.0f; b1 = 0.0f; c1 = -1.0f; }
  cW1[f * 4 + 0] = a1; cW1[f * 4 + 1] = b1; cW1[f * 4 + 2] = c1; cW1[f * 4 + 3] = 0.0f;
  cW2[f * 4 + 0] = a2; cW2[f * 4 + 1] = b2; cW2[f * 4 + 2] = c2; cW2[f * 4 + 3] = 0.0f;
  cD [f * 4 + 0] = ad; cD [f * 4 + 1] = bd; cD [f * 4 + 2] = cd; cD [f * 4 + 3] = 0.0f;
}

// ---------------------------------------------------------------------------
// Kernel 2: one wave32 per 16-pixel tile. For each 16-face tile, a single
// V_WMMA_F32_16X16X4_F32 evaluates one affine quantity for all 256
// (pixel,face) pairs. VALU does a branchless z-min (cndmask selects), which
// co-executes with the WMMAs on the matrix pipe.
// ---------------------------------------------------------------------------
__global__ void __launch_bounds__(256) render(const float* __restrict__ cW1,
                                              const float* __restrict__ cW2,
                                              const float* __restrict__ cD,
                                              const float* __restrict__ tex,
                                              const float* __restrict__ ref,
                                              float* __restrict__ partials) {
  const int lane = threadIdx.x & 31;
  const int wave = threadIdx.x >> 5;
  const int tile = blockIdx.x * 8 + wave;      // 0..1023
  const int row  = tile >> 3;                  // W/16 == 8 tiles per row
  const int col0 = (tile & 7) << 4;
  const int hf   = lane >> 4;                  // lane half: selects K pair
  const int ln   = lane & 15;

  // A-operand (16x4, M=pixel, K=(px,py,1,0)) per ISA VGPR layout.
  float px = 2.0f * ((float)(col0 + ln) + 0.5f) / (float)IMG_W - 1.0f;
  float py = 1.0f - 2.0f * ((float)row + 0.5f) / (float)IMG_H;
  v2f A;
  A.x = hf ? 1.0f : px;
  A.y = hf ? 0.0f : py;

  // bestz starts at FAR so (d < bestz) subsumes the (d < FAR) test.
  float bestz[8], bw1[8], bw2[8];
  int   bfid[8];
#pragma unroll
  for (int r = 0; r < 8; ++r) {
    bestz[r] = FAR_P; bw1[r] = 0.0f; bw2[r] = 0.0f; bfid[r] = 0;
  }

  const v8f zero = {};
  for (int ft = 0; ft < NFACE / 16; ++ft) {
    int face = ft * 16 + ln;                   // this lane's face column N
    int off  = face * 4 + hf * 2;
    v2f B1 = *(const v2f*)(cW1 + off);
    v2f B2 = *(const v2f*)(cW2 + off);
    v2f Bd = *(const v2f*)(cD  + off);
    v8f W1 = __builtin_amdgcn_wmma_f32_16x16x4_f32(false, A, false, B1,
                                                   (short)0, zero, false, false);
    v8f W2 = __builtin_amdgcn_wmma_f32_16x16x4_f32(false, A, false, B2,
                                                   (short)0, zero, false, false);
    v8f Dp = __builtin_amdgcn_wmma_f32_16x16x4_f32(false, A, false, Bd,
                                                   (short)0, zero, false, false);
#pragma unroll
    for (int r = 0; r < 8; ++r) {
      float w1 = W1[r], w2 = W2[r], d = Dp[r];
      float w0 = 1.0f - w1 - w2;
      float wmin = fminf(fminf(w0, w1), w2);   // v_min3_num_f32
      // strict < keeps earliest face on exact depth ties
      bool upd = (wmin >= 0.0f) & (d > NEAR_P) & (d < bestz[r]);
      bestz[r] = upd ? d    : bestz[r];        // v_cndmask chain, no branches
      bw1[r]   = upd ? w1   : bw1[r];
      bw2[r]   = upd ? w2   : bw2[r];
      bfid[r]  = upd ? face : bfid[r];
    }
  }

  // Cross-lane (z, fid) lexicographic min within each 16-lane half;
  // lane (hf, ln==r) keeps pixel M = r + 8*hf.
  float myz = FAR_P, myw1 = 0.0f, myw2 = 0.0f;
  int myfid = 0;
#pragma unroll
  for (int r = 0; r < 8; ++r) {
    float z = bestz[r], w1 = bw1[r], w2 = bw2[r];
    int fd = bfid[r];
#pragma unroll
    for (int m = 1; m < 16; m <<= 1) {
      float oz = __shfl_xor(z,  m, 32);
      float o1 = __shfl_xor(w1, m, 32);
      float o2 = __shfl_xor(w2, m, 32);
      int   of = __shfl_xor(fd, m, 32);
      bool take = (oz < z) || ((oz == z) && (of < fd));
      z  = take ? oz : z;
      w1 = take ? o1 : w1;
      w2 = take ? o2 : w2;
      fd = take ? of : fd;
    }
    if (ln == r) { myz = z; myw1 = w1; myw2 = w2; myfid = fd; }
  }

  // Trilinear sample of tanh(texture) + squared error (lanes with ln<8 own
  // one pixel each; others contribute 0).
  float err = 0.0f;
  if (ln < 8) {
    bool hit = myz < FAR_P;
    float col[3] = {0.0f, 0.0f, 0.0f};
    if (hit) {
      float w1 = myw1, w2 = myw2, w0 = 1.0f - w1 - w2;
      float cc[3] = {w0, w1, w2};
      int lo[3]; float fr[3];
#pragma unroll
      for (int i = 0; i < 3; ++i) {
        float v = fminf(fmaxf(cc[i], 0.0f), 1.0f) * (float)(TSZ - 1);
        int l = (int)floorf(v);
        l = l < 0 ? 0 : (l > TSZ - 2 ? TSZ - 2 : l);
        lo[i] = l; fr[i] = v - (float)l;
      }
      const float* tb = tex + myfid * (TSZ * TSZ * TSZ * 3);
#pragma unroll
      for (int a = 0; a < 2; ++a)
#pragma unroll
        for (int b = 0; b < 2; ++b)
#pragma unroll
          for (int d = 0; d < 2; ++d) {
            float wgt = (a ? fr[0] : 1.0f - fr[0]) *
                        (b ? fr[1] : 1.0f - fr[1]) *
                        (d ? fr[2] : 1.0f - fr[2]);
            const float* tp = tb + ((lo[0] + a) * 16 + (lo[1] + b) * 4 + (lo[2] + d)) * 3;
            col[0] += wgt * tanhf(tp[0]);
            col[1] += wgt * tanhf(tp[1]);
            col[2] += wgt * tanhf(tp[2]);
          }
    }
    int p = row * IMG_W + col0 + (ln + 8 * hf);
#pragma unroll
    for (int ch = 0; ch < 3; ++ch) {
      float dlt = col[ch] - ref[ch * (IMG_H * IMG_W) + p];
      err += dlt * dlt;
    }
  }

  // Deterministic block reduction (wave shfl tree + LDS).
#pragma unroll
  for (int m = 16; m >= 1; m >>= 1) err += __shfl_xor(err, m, 32);
  __shared__ float ssum[8];
  if (lane == 0) ssum[wave] = err;
  __syncthreads();
  if (threadIdx.x == 0) {
    float s = 0.0f;
#pragma unroll
    for (int i = 0; i < 8; ++i) s += ssum[i];
    partials[blockIdx.x] = s;
  }
}

// Kernel 3: fixed-order tree sum of 128 block partials -> scalar loss.
__global__ void finish(const float* __restrict__ partials, float* __restrict__ out) {
  __shared__ float s[128];
  s[threadIdx.x] = partials[threadIdx.x];
  __syncthreads();
  for (int step = 64; step > 0; step >>= 1) {
    if (threadIdx.x < step) s[threadIdx.x] += s[threadIdx.x + step];
    __syncthreads();
  }
  if (threadIdx.x == 0) out[0] = s[0];
}

extern "C" void kernel_launch(void* const* d_in, const int* in_sizes, int n_in,
                              void* d_out, int out_size, void* d_ws, size_t ws_size,
                              hipStream_t stream) {
  (void)in_sizes; (void)n_in; (void)out_size; (void)ws_size;
  const float* verts = (const float*)d_in[0];   // (1,642,3)  f32
  const int*   faces = (const int*)d_in[1];     // (1,1280,3) i32
  const float* tex   = (const float*)d_in[2];   // (1,1280,4,4,4,3) f32
  const float* ref   = (const float*)d_in[3];   // (1,3,128,128) f32
  float* out = (float*)d_out;

  char* ws = (char*)d_ws;
  float* cW1 = (float*)(ws + 0);
  float* cW2 = (float*)(ws + 20480);
  float* cD  = (float*)(ws + 40960);
  float* partials = (float*)(ws + 61440);

  // Camera constants (replicates the reference's float32 math on host).
  const float dist = 2.732f;
  float az = 90.0f * (float)M_PI / 180.0f;
  float el = 0.0f;
  float eye[3] = { dist * cosf(el) * sinf(az),
                   dist * sinf(el),
                  -dist * cosf(el) * cosf(az) };
  float nz[3] = { -eye[0], -eye[1], -eye[2] };
  float nl = sqrtf(nz[0]*nz[0] + nz[1]*nz[1] + nz[2]*nz[2] + 1e-12f);
  float zax[3] = { nz[0]/nl, nz[1]/nl, nz[2]/nl };
  float up[3] = { 0.0f, 1.0f, 0.0f };
  float cx[3] = { up[1]*zax[2] - up[2]*zax[1],
                  up[2]*zax[0] - up[0]*zax[2],
                  up[0]*zax[1] - up[1]*zax[0] };
  float cl = sqrtf(cx[0]*cx[0] + cx[1]*cx[1] + cx[2]*cx[2] + 1e-12f);
  float xax[3] = { cx[0]/cl, cx[1]/cl, cx[2]/cl };
  float yax[3] = { zax[1]*xax[2] - zax[2]*xax[1],
                   zax[2]*xax[0] - zax[0]*xax[2],
                   zax[0]*xax[1] - zax[1]*xax[0] };

  face_setup<<<(NFACE + 255) / 256, 256, 0, stream>>>(
      verts, faces, cW1, cW2, cD,
      eye[0], eye[1], eye[2],
      xax[0], xax[1], xax[2],
      yax[0], yax[1], yax[2],
      zax[0], zax[1], zax[2]);

  render<<<128, 256, 0, stream>>>(cW1, cW2, cD, tex, ref, partials);

  finish<<<1, 128, 0, stream>>>(partials, out);
}